// LightGCNLayer_20031727468914
// MI455X (gfx1250) — compile-verified
//
#include <hip/hip_runtime.h>
#include <hip/hip_bf16.h>

#define TILE 256
#define DIM  64

__global__ __launch_bounds__(256) void zero_f32v4_kernel(float4* __restrict__ p, long long n4) {
    long long i = (long long)blockIdx.x * blockDim.x + threadIdx.x;
    if (i < n4) p[i] = make_float4(0.f, 0.f, 0.f, 0.f);
}

__global__ __launch_bounds__(256) void zero_u32_kernel(unsigned* __restrict__ p, long long n) {
    long long i = (long long)blockIdx.x * blockDim.x + threadIdx.x;
    if (i < n) p[i] = 0u;
}

// One thread per edge: integer degree counts (exact; matches segment_sum of ones).
__global__ __launch_bounds__(256) void degree_kernel(const int* __restrict__ uidx,
                                                     const int* __restrict__ iidx,
                                                     unsigned* __restrict__ udeg,
                                                     unsigned* __restrict__ ideg,
                                                     int E) {
    int e = blockIdx.x * 256 + threadIdx.x;
    if (e < E) {
        atomicAdd(&udeg[uidx[e]], 1u);
        atomicAdd(&ideg[iidx[e]], 1u);
    }
}

// One block = one tile of 256 edges.
// Phase 1: edge metadata (u, i, w) staged into LDS via CDNA5 async global->LDS
//          loads (ASYNCcnt), drained with s_wait_asynccnt.
// Phase 2: each thread normalizes ITS OWN staged edge once:
//          s_w[t] = w * rsqrt(clip(du,1)*clip(di,1))   (1 gather+rsqrt per edge,
//          not 16 -- removes 15/16 of degree traffic and transcendentals).
// Phase 3: 16 lanes fan out per edge; lane 'sub' owns float4 chunk of the
//          64-dim row -> coalesced 256B row reads, hardware f32 atomics out.
__global__ __launch_bounds__(256) void scatter_kernel(const float* __restrict__ u_emb,
                                                      const float* __restrict__ i_emb,
                                                      const int* __restrict__ uidx,
                                                      const int* __restrict__ iidx,
                                                      const float* __restrict__ wts,
                                                      const unsigned* __restrict__ udeg,
                                                      const unsigned* __restrict__ ideg,
                                                      float* __restrict__ new_u,
                                                      float* __restrict__ new_i,
                                                      int E) {
    __shared__ int   s_u[TILE];
    __shared__ int   s_i[TILE];
    __shared__ float s_w[TILE];

    const int tid  = threadIdx.x;
    const int base = blockIdx.x * TILE;

    // ---- phase 1: async stage (clamped for the tail tile) ----
    {
        int e = base + tid;
        unsigned boff = (unsigned)((e < E) ? e : (E - 1)) * 4u;
        // LDS byte addresses: low 32 bits of the generic pointer are the LDS offset.
        unsigned a_u = (unsigned)(uintptr_t)(&s_u[tid]);
        unsigned a_i = (unsigned)(uintptr_t)(&s_i[tid]);
        unsigned a_w = (unsigned)(uintptr_t)(&s_w[tid]);
        asm volatile("global_load_async_to_lds_b32 %0, %1, %2"
                     :: "v"(a_u), "v"(boff), "s"(uidx) : "memory");
        asm volatile("global_load_async_to_lds_b32 %0, %1, %2"
                     :: "v"(a_i), "v"(boff), "s"(iidx) : "memory");
        asm volatile("global_load_async_to_lds_b32 %0, %1, %2"
                     :: "v"(a_w), "v"(boff), "s"(wts) : "memory");
        asm volatile("s_wait_asynccnt 0" ::: "memory");
    }
    __syncthreads();

    // ---- phase 2: per-edge symmetric degree normalization (once per edge) ----
    {
        const int uu = s_u[tid];
        const int ii = s_i[tid];
        unsigned du = udeg[uu]; du = (du < 1u) ? 1u : du;   // clip(deg, 1)
        unsigned di = ideg[ii]; di = (di < 1u) ? 1u : di;
        s_w[tid] = s_w[tid] * rsqrtf((float)du * (float)di);
    }
    __syncthreads();

    // ---- phase 3: 16-lane fanout per edge ----
    const int sub = tid & 15;   // float4 column chunk within the 64-dim row
    const int grp = tid >> 4;   // 16 edge-groups per block

    #pragma unroll 4
    for (int j = 0; j < 16; ++j) {
        const int el = grp * 16 + j;      // 0..255 within tile
        const int e  = base + el;
        if (e >= E) break;

        const int   uu = s_u[el];
        const int   ii = s_i[el];
        const float we = s_w[el];

        const float4 iv = *(const float4*)(i_emb + (size_t)ii * DIM + sub * 4);
        const float4 uv = *(const float4*)(u_emb + (size_t)uu * DIM + sub * 4);

        float* pu = new_u + (size_t)uu * DIM + sub * 4;
        unsafeAtomicAdd(pu + 0, iv.x * we);
        unsafeAtomicAdd(pu + 1, iv.y * we);
        unsafeAtomicAdd(pu + 2, iv.z * we);
        unsafeAtomicAdd(pu + 3, iv.w * we);

        float* pi = new_i + (size_t)ii * DIM + sub * 4;
        unsafeAtomicAdd(pi + 0, uv.x * we);
        unsafeAtomicAdd(pi + 1, uv.y * we);
        unsafeAtomicAdd(pi + 2, uv.z * we);
        unsafeAtomicAdd(pi + 3, uv.w * we);
    }
}

extern "C" void kernel_launch(void* const* d_in, const int* in_sizes, int n_in,
                              void* d_out, int out_size, void* d_ws, size_t ws_size,
                              hipStream_t stream) {
    const float* u_emb = (const float*)d_in[0];
    const float* i_emb = (const float*)d_in[1];
    const int*   edges = (const int*)d_in[2];
    const float* wts   = (const float*)d_in[3];

    const int NU = in_sizes[0] / DIM;   // 100000
    const int NI = in_sizes[1] / DIM;   // 50000
    const int E  = in_sizes[3];         // 4000000

    const int* uidx = edges;
    const int* iidx = edges + E;

    float* new_u = (float*)d_out;
    float* new_i = new_u + (size_t)NU * DIM;

    unsigned* udeg = (unsigned*)d_ws;
    unsigned* ideg = udeg + NU;

    // 1) zero outputs (d_out is poisoned) and degree scratch.
    //    (NU+NI)*DIM is a multiple of 4 -> float4 stores are exact.
    {
        long long n4 = (long long)(NU + NI) * DIM / 4;
        zero_f32v4_kernel<<<(unsigned)((n4 + 255) / 256), 256, 0, stream>>>((float4*)d_out, n4);
        long long n_deg = (long long)NU + NI;
        zero_u32_kernel<<<(unsigned)((n_deg + 255) / 256), 256, 0, stream>>>(udeg, n_deg);
    }

    // 2) degrees
    degree_kernel<<<(E + 255) / 256, 256, 0, stream>>>(uidx, iidx, udeg, ideg, E);

    // 3) normalized scatter of both embedding directions
    scatter_kernel<<<(E + TILE - 1) / TILE, 256, 0, stream>>>(
        u_emb, i_emb, uidx, iidx, wts, udeg, ideg, new_u, new_i, E);
}